// SSIM_Loss_80917183856973
// MI455X (gfx1250) — compile-verified
//
#include <hip/hip_runtime.h>
#include <hip/hip_bf16.h>
#include <math.h>

// ---------------------------------------------------------------------------
// SSIM loss on MI455X (gfx1250), f32-exact via V_WMMA_F32_16X16X4_F32.
//
// Separable Gaussian: 2D kernel K = g (x) g, with g[i] = sqrt(K[i][i]).
// Each wave32 computes one 16x16 output tile:
//   horizontal pass:  H(38x16)  = P(38x40) * T(40x16),  T[k][c] = g[k-c]
//   vertical   pass:  D(16x16)  = T'(16x40) * H(40x16), T'[m][k] = g[k-m]
// Both Toeplitz operands share one per-lane register table tw[10] (v2f).
// 5 fields (a, t, a^2, t^2, a*t) share the raw-patch LDS loads.
// ---------------------------------------------------------------------------

typedef float v2f __attribute__((ext_vector_type(2)));
typedef float v8f __attribute__((ext_vector_type(8)));

#define IMG_H   512
#define IMG_W   512
#define N_IMG   48            // 16 batch * 3 channels
#define KS      23            // gaussian taps
#define PAD     11
#define SSIM_C1 1.0e-4f
#define SSIM_C2 9.0e-4f

#define TILES_X 32
#define TILES_Y 32
#define N_TILES (N_IMG * TILES_X * TILES_Y)   // 49152
#define WPB     2             // waves per block (64 threads)
#define PATCH   38            // 16 + 2*11 halo
#define KPAD    40            // K padded to 10 chunks of 4
#define PSTR    41            // LDS row stride (bank spread)

__global__ void ssim_zero_kernel(float* acc) {
    if (threadIdx.x == 0 && blockIdx.x == 0) acc[0] = 0.0f;
}

__global__ void ssim_finalize_kernel(const float* acc, float* out) {
    if (threadIdx.x == 0 && blockIdx.x == 0)
        out[0] = 1.0f - acc[0] * (1.0f / (float)((size_t)N_IMG * IMG_H * IMG_W));
}

__global__ __launch_bounds__(32 * WPB)
void ssim_tile_kernel(const float* __restrict__ gin,
                      const float* __restrict__ gtg,
                      const float* __restrict__ gwt,
                      float* __restrict__ accum) {
    __shared__ float s_a[WPB][PATCH * PSTR];   // raw input patch per wave
    __shared__ float s_t[WPB][PATCH * PSTR];   // raw target patch per wave
    __shared__ float s_H[WPB][5][KPAD * 16];   // horizontal-pass result per field
    __shared__ float s_g[KS];                  // 1-D gaussian taps

    const int tid  = threadIdx.x;
    const int wave = tid >> 5;
    const int lane = tid & 31;
    const int half = lane >> 4;     // lane-half selects K offset (+2) in A/B frags
    const int j    = lane & 15;     // M row (A) / N col (B,C,D)

    // Recover the exact separable 1-D kernel from the 2-D weight (channel 0).
    if (tid == 0) {
        for (int i = 0; i < KS; ++i) s_g[i] = sqrtf(gwt[i * KS + i]);
    }

    const int tileIdx = blockIdx.x * WPB + wave;      // grid sized exactly
    const int img  = tileIdx >> 10;                   // / (32*32)
    const int trem = tileIdx & 1023;
    const int y0 = (trem >> 5) << 4;
    const int x0 = (trem & 31) << 4;

    const float* imA = gin + (size_t)img * IMG_H * IMG_W;
    const float* imT = gtg + (size_t)img * IMG_H * IMG_W;
    float* pa = s_a[wave];
    float* pt = s_t[wave];

    // Cooperative zero-padded patch load (wave-private LDS).
    for (int i = lane; i < PATCH * KPAD; i += 32) {
        const int r = i / KPAD, c = i - r * KPAD;
        const int gy = y0 - PAD + r, gx = x0 - PAD + c;
        const bool ok = (c < PATCH) & (gy >= 0) & (gy < IMG_H) & (gx >= 0) & (gx < IMG_W);
        pa[r * PSTR + c] = ok ? imA[gy * IMG_W + gx] : 0.0f;
        pt[r * PSTR + c] = ok ? imT[gy * IMG_W + gx] : 0.0f;
    }
    __syncthreads();   // publishes s_g (patches are wave-private / in-order DS)

    // Toeplitz fragments, shared by both passes (index by lane%16 vs K).
    v2f tw[10];
#pragma unroll
    for (int q = 0; q < 10; ++q) {
        const int k0 = 4 * q + 2 * half;
        const int i0 = k0 - j, i1 = k0 + 1 - j;
        tw[q].x = (i0 >= 0 && i0 < KS) ? s_g[i0] : 0.0f;
        tw[q].y = (i1 >= 0 && i1 < KS) ? s_g[i1] : 0.0f;
    }

    // ---------------- horizontal pass: 3 row-chunks x 10 K-steps x 5 fields
#pragma unroll
    for (int mc = 0; mc < 3; ++mc) {
        v8f acc0 = {}, acc1 = {}, acc2 = {}, acc3 = {}, acc4 = {};
        const int  prow = 16 * mc + j;           // A-frag: lane%16 = M = patch row
        const bool rok  = prow < PATCH;
        const float* ra = pa + prow * PSTR;
        const float* rt = pt + prow * PSTR;
#pragma unroll
        for (int q = 0; q < 10; ++q) {
            const int k0 = 4 * q + 2 * half;     // A-frag K: VGPR comp + lane-half
            const float a0 = rok ? ra[k0]     : 0.0f;
            const float a1 = rok ? ra[k0 + 1] : 0.0f;
            const float t0 = rok ? rt[k0]     : 0.0f;
            const float t1 = rok ? rt[k0 + 1] : 0.0f;
            const v2f Am1 = {a0, a1};
            const v2f Am2 = {t0, t1};
            const v2f Aaa = {a0 * a0, a1 * a1};
            const v2f Att = {t0 * t0, t1 * t1};
            const v2f Aat = {a0 * t0, a1 * t1};
            acc0 = __builtin_amdgcn_wmma_f32_16x16x4_f32(false, Am1, false, tw[q], (short)0, acc0, false, false);
            acc1 = __builtin_amdgcn_wmma_f32_16x16x4_f32(false, Am2, false, tw[q], (short)0, acc1, false, false);
            acc2 = __builtin_amdgcn_wmma_f32_16x16x4_f32(false, Aaa, false, tw[q], (short)0, acc2, false, false);
            acc3 = __builtin_amdgcn_wmma_f32_16x16x4_f32(false, Att, false, tw[q], (short)0, acc3, false, false);
            acc4 = __builtin_amdgcn_wmma_f32_16x16x4_f32(false, Aat, false, tw[q], (short)0, acc4, false, false);
        }
        // D layout: comp g -> row 16*mc + g + 8*half, col lane%16.
#pragma unroll
        for (int g = 0; g < 8; ++g) {
            const int Mr = 16 * mc + g + 8 * half;
            if (Mr < KPAD) {
                const int off = Mr * 16 + j;
                s_H[wave][0][off] = acc0[g];
                s_H[wave][1][off] = acc1[g];
                s_H[wave][2][off] = acc2[g];
                s_H[wave][3][off] = acc3[g];
                s_H[wave][4][off] = acc4[g];
            }
        }
    }
    // DS ops are in-order within a wave: safe to read s_H[wave] now.

    // ---------------- vertical pass: 10 K-steps per field
    v8f D[5];
#pragma unroll
    for (int f = 0; f < 5; ++f) {
        v8f acc = {};
        const float* Hf = &s_H[wave][f][0];
#pragma unroll
        for (int q = 0; q < 10; ++q) {
            const int k0 = 4 * q + 2 * half;
            v2f B;
            B.x = Hf[k0 * 16 + j];               // B-frag: lane%16 = N, comp/half = K
            B.y = Hf[(k0 + 1) * 16 + j];
            acc = __builtin_amdgcn_wmma_f32_16x16x4_f32(false, tw[q], false, B, (short)0, acc, false, false);
        }
        D[f] = acc;
    }

    // ---------------- SSIM map + reduction (8 pixels per lane)
    float sum = 0.0f;
#pragma unroll
    for (int g = 0; g < 8; ++g) {
        const float m1 = D[0][g], m2 = D[1][g];
        const float m1s = m1 * m1, m2s = m2 * m2, m12 = m1 * m2;
        const float sg1 = D[2][g] - m1s;
        const float sg2 = D[3][g] - m2s;
        const float s12 = D[4][g] - m12;
        const float num = (2.0f * m12 + SSIM_C1) * (2.0f * s12 + SSIM_C2);
        const float den = (m1s + m2s + SSIM_C1) * (sg1 + sg2 + SSIM_C2);
        sum += num / den;
    }
#pragma unroll
    for (int off = 16; off > 0; off >>= 1) sum += __shfl_xor(sum, off, 32);
    if (lane == 0) atomicAdd(accum, sum);
}

extern "C" void kernel_launch(void* const* d_in, const int* in_sizes, int n_in,
                              void* d_out, int out_size, void* d_ws, size_t ws_size,
                              hipStream_t stream) {
    (void)in_sizes; (void)n_in; (void)out_size; (void)ws_size;
    const float* gin = (const float*)d_in[0];
    const float* gtg = (const float*)d_in[1];
    const float* gwt = (const float*)d_in[2];
    float* acc = (float*)d_ws;
    float* out = (float*)d_out;

    ssim_zero_kernel<<<1, 32, 0, stream>>>(acc);
    ssim_tile_kernel<<<N_TILES / WPB, 32 * WPB, 0, stream>>>(gin, gtg, gwt, acc);
    ssim_finalize_kernel<<<1, 32, 0, stream>>>(acc, out);
}